// LightGCN_Hetero_7103875908136
// MI455X (gfx1250) — compile-verified
//
#include <hip/hip_runtime.h>
#include <hip/hip_bf16.h>
#include <stdint.h>

#define NUM_USERS  100000
#define NUM_ITEMS  150000
#define NUM_BRANDS 5000
#define NUM_NODES  (NUM_USERS + NUM_ITEMS + NUM_BRANDS)   // 255000
#define EMBED_DIM  64
#define N_LAYERS   3
#define NUM_EDGES  2000000
#define OUT_ROWS   (NUM_USERS + NUM_ITEMS)                // 250000

// float4 element counts
#define EGO_F4 (NUM_NODES * EMBED_DIM / 4)   // 4,080,000
#define OUT_F4 (OUT_ROWS  * EMBED_DIM / 4)   // 4,000,000

typedef __attribute__((ext_vector_type(2))) float v2f;
typedef __attribute__((ext_vector_type(8))) float v8f;

// ---------------------------------------------------------------------------
// ego = concat(user, item, brand); out (acc) initialized to ego[user+item]
// ---------------------------------------------------------------------------
__global__ void init_k(const float4* __restrict__ u,
                       const float4* __restrict__ it,
                       const float4* __restrict__ br,
                       float4* __restrict__ ego,
                       float4* __restrict__ out) {
  unsigned i = blockIdx.x * blockDim.x + threadIdx.x;
  if (i >= EGO_F4) return;
  const unsigned NU4 = NUM_USERS * EMBED_DIM / 4;   // 1,600,000
  const unsigned NI4 = NUM_ITEMS * EMBED_DIM / 4;   // 2,400,000
  float4 v;
  if (i < NU4)            v = u[i];
  else if (i < NU4 + NI4) v = it[i - NU4];
  else                    v = br[i - NU4 - NI4];
  ego[i] = v;
  if (i < OUT_F4) out[i] = v;   // acc starts as ego (user+item rows)
}

__global__ void zero_k(float4* __restrict__ p) {
  unsigned i = blockIdx.x * blockDim.x + threadIdx.x;
  if (i < EGO_F4) p[i] = make_float4(0.f, 0.f, 0.f, 0.f);
}

__global__ void acc_k(float4* __restrict__ out, const float4* __restrict__ nxt) {
  unsigned i = blockIdx.x * blockDim.x + threadIdx.x;
  if (i >= OUT_F4) return;
  float4 a = out[i], b = nxt[i];
  a.x += b.x; a.y += b.y; a.z += b.z; a.w += b.w;
  out[i] = a;
}

// ---------------------------------------------------------------------------
// SpMM: nxt[row] += val * ego[col] for each COO edge.
// Edge triples are staged into LDS with CDNA5 async global->LDS loads
// (ASYNCcnt / s_wait_asynccnt). Each wave stages and consumes its own
// 32-edge slice, so no workgroup barrier is required. Each wave then covers
// one edge's 64-float row as 32 x float2 (coalesced 256B gather from the
// L2-resident table) and scatters with hardware f32 atomics.
// ---------------------------------------------------------------------------
__global__ void spmm_k(const int* __restrict__ rows,
                       const int* __restrict__ cols,
                       const float* __restrict__ vals,
                       const float* __restrict__ ego,
                       float* __restrict__ nxt) {
  __shared__ int   s_rows[256];
  __shared__ int   s_cols[256];
  __shared__ float s_vals[256];

  const unsigned lane = threadIdx.x & 31u;
  const unsigned wv   = threadIdx.x >> 5;
  const unsigned slot = threadIdx.x;                 // one staged edge per thread
  const unsigned e    = blockIdx.x * 256u + slot;
  const unsigned ec   = e < NUM_EDGES ? e : (NUM_EDGES - 1u);
  const unsigned voff = ec * 4u;                     // byte offset into edge arrays

  unsigned lds_r = (unsigned)(uintptr_t)&s_rows[slot];
  unsigned lds_c = (unsigned)(uintptr_t)&s_cols[slot];
  unsigned lds_v = (unsigned)(uintptr_t)&s_vals[slot];
  uint64_t base_r = (uint64_t)rows;
  uint64_t base_c = (uint64_t)cols;
  uint64_t base_v = (uint64_t)vals;

  asm volatile("global_load_async_to_lds_b32 %0, %1, %2 offset:0"
               :: "v"(lds_r), "v"(voff), "s"(base_r) : "memory");
  asm volatile("global_load_async_to_lds_b32 %0, %1, %2 offset:0"
               :: "v"(lds_c), "v"(voff), "s"(base_c) : "memory");
  asm volatile("global_load_async_to_lds_b32 %0, %1, %2 offset:0"
               :: "v"(lds_v), "v"(voff), "s"(base_v) : "memory");
  asm volatile("s_wait_asynccnt 0" ::: "memory");

  const unsigned base_w = blockIdx.x * 256u + wv * 32u;
  #pragma unroll 4
  for (unsigned j = 0; j < 32u; ++j) {
    if (base_w + j >= NUM_EDGES) break;              // wave-uniform
    int   row = s_rows[wv * 32u + j];
    int   col = s_cols[wv * 32u + j];
    float val = s_vals[wv * 32u + j];
    const float2 x = *(const float2*)(ego + (size_t)col * EMBED_DIM + 2u * lane);
    float* dst = nxt + (size_t)row * EMBED_DIM + 2u * lane;
    __hip_atomic_fetch_add(dst,     val * x.x, __ATOMIC_RELAXED, __HIP_MEMORY_SCOPE_AGENT);
    __hip_atomic_fetch_add(dst + 1, val * x.y, __ATOMIC_RELAXED, __HIP_MEMORY_SCOPE_AGENT);
  }
}

// ---------------------------------------------------------------------------
// final scale: out = acc * 0.25 via V_WMMA_F32_16X16X4_F32.
// One wave owns a 16x16 tile. D = sum_j A_j(16x4) * B_j(4x16) with
// A_j = 0.25 * selector of rows 4j..4j+3 -> D = 0.25 * X.
// VGPR layouts per CDNA5 ISA 7.12.2:
//   A 16x4: vgpr0 = K{0|2}, vgpr1 = K{1|3} (low/high half-wave)
//   B 4x16: vgpr0 = row K{0|2}, vgpr1 = row K{1|3}, N = lane&15
//   C/D   : vgpr v = row v (lanes 0-15) / row v+8 (lanes 16-31)
// ---------------------------------------------------------------------------
__global__ void finalize_wmma_k(float* __restrict__ out) {
  const unsigned ROW_TILES = OUT_ROWS / 16;          // 15625
  const unsigned COL_TILES = EMBED_DIM / 16;         // 4
  unsigned wave = blockIdx.x * (blockDim.x >> 5) + (threadIdx.x >> 5);
  if (wave >= ROW_TILES * COL_TILES) return;         // wave-uniform; EXEC stays all-1s

  const unsigned row0 = (wave >> 2) * 16u;
  const unsigned col0 = (wave & 3u) * 16u;
  const unsigned lane = threadIdx.x & 31u;
  const unsigned half = lane >> 4;                   // 0: lanes 0-15, 1: lanes 16-31
  const unsigned l    = lane & 15u;
  const unsigned k0   = 2u * half;                   // K held in vgpr0
  const unsigned k1   = 2u * half + 1u;              // K held in vgpr1

  v8f c = {};
  #pragma unroll
  for (unsigned j = 0; j < 4u; ++j) {
    v2f a, b;
    a.x = (l == 4u * j + k0) ? 0.25f : 0.0f;
    a.y = (l == 4u * j + k1) ? 0.25f : 0.0f;
    b.x = out[(size_t)(row0 + 4u * j + k0) * EMBED_DIM + col0 + l];
    b.y = out[(size_t)(row0 + 4u * j + k1) * EMBED_DIM + col0 + l];
    c = __builtin_amdgcn_wmma_f32_16x16x4_f32(false, a, false, b,
                                              (short)0, c, false, false);
  }
  #pragma unroll
  for (unsigned v = 0; v < 8u; ++v)
    out[(size_t)(row0 + v + 8u * half) * EMBED_DIM + col0 + l] = c[v];
}

// ---------------------------------------------------------------------------
extern "C" void kernel_launch(void* const* d_in, const int* in_sizes, int n_in,
                              void* d_out, int out_size, void* d_ws, size_t ws_size,
                              hipStream_t stream) {
  const float* user  = (const float*)d_in[0];
  const float* item  = (const float*)d_in[1];
  const float* brand = (const float*)d_in[2];
  const int*   rows  = (const int*)d_in[3];
  const int*   cols  = (const int*)d_in[4];
  const float* vals  = (const float*)d_in[5];
  float* out = (float*)d_out;

  float* buf0 = (float*)d_ws;
  float* buf1 = buf0 + (size_t)NUM_NODES * EMBED_DIM;   // 65.28 MB each

  const unsigned TB = 256;
  init_k<<<(EGO_F4 + TB - 1) / TB, TB, 0, stream>>>(
      (const float4*)user, (const float4*)item, (const float4*)brand,
      (float4*)buf0, (float4*)out);

  float* ego = buf0;
  float* nxt = buf1;
  for (int layer = 0; layer < N_LAYERS; ++layer) {
    zero_k<<<(EGO_F4 + TB - 1) / TB, TB, 0, stream>>>((float4*)nxt);
    spmm_k<<<(NUM_EDGES + TB - 1) / TB, TB, 0, stream>>>(rows, cols, vals, ego, nxt);
    acc_k<<<(OUT_F4 + TB - 1) / TB, TB, 0, stream>>>((float4*)out, (const float4*)nxt);
    float* t = ego; ego = nxt; nxt = t;
  }

  const unsigned tiles = (OUT_ROWS / 16) * (EMBED_DIM / 16);   // 62500 waves
  finalize_wmma_k<<<(tiles + 7) / 8, TB, 0, stream>>>(out);
}